// GaussianNoise_35510789603412
// MI455X (gfx1250) — compile-verified
//
#include <hip/hip_runtime.h>
#include <hip/hip_bf16.h>

// out = noised + (2*STD) * noise,  STD = 0.05  ->  scale = 0.1
// Pure streaming AXPY over 50,331,648 fp32 elems: 12 B/elem, ~604 MB/call.
// HBM-bound: floor ~26 us at 23.3 TB/s. Compute is irrelevant (0.083 FLOP/B),
// so no WMMA — the optimal CDNA5 path is b128 non-temporal VMEM + packed FMA.
//
// This revision: 32-bit indexing (max byte offset 201 MB < 2^31) so the
// backend can use scalar-base + 32-bit VGPR-offset global addressing and drop
// the 64-bit address adds; one float4 per thread, no loop machinery.

typedef float v4f __attribute__((ext_vector_type(4)));

__global__ __launch_bounds__(256) void GaussianNoise_axpy_v4(
    const v4f* __restrict__ xv,   // noised, as float4
    const v4f* __restrict__ nv,   // noise, as float4
    v4f* __restrict__ ov,         // out, as float4
    int n4)                       // number of float4 elements
{
    int i = (int)(blockIdx.x * blockDim.x + threadIdx.x);
    if (i < n4) {
        // global_load_b128 th:TH_LOAD_NT (stream, don't pollute L2)
        v4f a = __builtin_nontemporal_load(xv + i);
        v4f b = __builtin_nontemporal_load(nv + i);
        v4f c = a + 0.1f * b;     // 2x v_pk_fma_f32
        // global_store_b128 th:TH_STORE_NT
        __builtin_nontemporal_store(c, ov + i);
    }
}

// Scalar tail for n % 4 != 0 (not used for this problem size, but correct).
__global__ __launch_bounds__(64) void GaussianNoise_axpy_tail(
    const float* __restrict__ x,
    const float* __restrict__ nse,
    float* __restrict__ out,
    int start, int n)
{
    int i = start + (int)(blockIdx.x * blockDim.x + threadIdx.x);
    if (i < n) {
        out[i] = fmaf(0.1f, nse[i], x[i]);
    }
}

extern "C" void kernel_launch(void* const* d_in, const int* in_sizes, int n_in,
                              void* d_out, int out_size, void* d_ws, size_t ws_size,
                              hipStream_t stream) {
    const float* noised = (const float*)d_in[0];
    const float* noise  = (const float*)d_in[1];
    float* out          = (float*)d_out;

    const int n   = out_size;        // 50,331,648
    const int n4  = n >> 2;          // 12,582,912 float4s
    const int rem = n & 3;           // 0 for this problem

    if (n4 > 0) {
        const int block = 256;       // 8 wave32s
        const int grid  = (n4 + block - 1) / block;   // ~49,152 blocks
        GaussianNoise_axpy_v4<<<grid, block, 0, stream>>>(
            (const v4f*)noised, (const v4f*)noise, (v4f*)out, n4);
    }
    if (rem != 0) {
        GaussianNoise_axpy_tail<<<1, 64, 0, stream>>>(
            noised, noise, out, n4 << 2, n);
    }
}